// MUSIC_80126909874218
// MI455X (gfx1250) — compile-verified
//
#include <hip/hip_runtime.h>
#include <math.h>

typedef __attribute__((ext_vector_type(2))) float v2f;
typedef __attribute__((ext_vector_type(8))) float v8f;

#define PI_F 3.14159265358979323846f
#define NFFT 512
#define HOPSZ 256
#define NFREQ 257
#define NMICS 4
#define ESTN 4
#define NANG 361
#define TSTART 120
#define MEPS 1e-8f
// 2*pi*D_INTER/C_SOUND
#define KPH (2.0f * PI_F * 0.04f / 343.0f)
// f_analog step = (FS/2)/(NFFT/2) = 8000/256
#define FSTEP 31.25f

// ---------------- Kernel 1: 16 windowed 512-pt DFTs (only frames 120..123) ----
__global__ void music_stft_kernel(const float* __restrict__ x, float* __restrict__ Xs) {
    int id = blockIdx.x * blockDim.x + threadIdx.x;
    if (id >= NFREQ * 16) return;
    int fi = id >> 4;
    int mt = id & 15;
    int m  = mt >> 2;
    int t  = mt & 3;
    // frame (120+t) starts at padded index (120+t)*256; pad=256 -> x index:
    const float* xp = x + (size_t)((TSTART * HOPSZ - NFFT / 2) + HOPSZ * t) * NMICS + m;
    float ar = 0.f, ai = 0.f;
    for (int n = 0; n < NFFT; ++n) {
        // sqrt-hann periodic window == sin(pi*n/N)
        float w = sinf((float)n * (PI_F / (float)NFFT));
        float v = xp[(size_t)n * NMICS] * w;
        int ph = (fi * n) & (NFFT - 1);          // exact phase mod 512
        float ang = (float)ph * (2.0f * PI_F / (float)NFFT);
        float s, c;
        sincosf(ang, &s, &c);
        ar = fmaf(v, c, ar);
        ai = fmaf(-v, s, ai);
    }
    float* o = Xs + ((size_t)(m * NFREQ + fi) * ESTN + t) * 2;
    o[0] = ar; o[1] = ai;
}

// ---------------- Kernel 2: per-freq 4x4 Hermitian eigh (complex Jacobi) -------
__global__ void music_eig_kernel(const float* __restrict__ Xs, float* __restrict__ noise) {
    int f = blockIdx.x * blockDim.x + threadIdx.x;
    if (f >= NFREQ) return;

    float xr[4][4], xi2[4][4];   // [mic][frame]
#pragma unroll
    for (int m = 0; m < 4; ++m)
#pragma unroll
        for (int t = 0; t < 4; ++t) {
            const float* p = Xs + ((size_t)(m * NFREQ + f) * ESTN + t) * 2;
            xr[m][t] = p[0];
            xi2[m][t] = p[1];
        }

    float Ar[4][4], Ai[4][4], Vr[4][4], Vi[4][4];
#pragma unroll
    for (int i = 0; i < 4; ++i)
#pragma unroll
        for (int j = 0; j < 4; ++j) {
            float sr = 0.f, si = 0.f;
#pragma unroll
            for (int t = 0; t < 4; ++t) {
                sr += xr[i][t] * xr[j][t] + xi2[i][t] * xi2[j][t];
                si += xi2[i][t] * xr[j][t] - xr[i][t] * xi2[j][t];
            }
            Ar[i][j] = 0.25f * sr;
            Ai[i][j] = 0.25f * si;
            Vr[i][j] = (i == j) ? 1.f : 0.f;
            Vi[i][j] = 0.f;
        }

    // cyclic complex Jacobi, 12 sweeps (4x4: converges to machine precision)
    for (int sweep = 0; sweep < 12; ++sweep) {
#pragma unroll
        for (int k = 0; k < 6; ++k) {
            const int PP[6] = {0, 0, 0, 1, 1, 2};
            const int QQ[6] = {1, 2, 3, 2, 3, 3};
            int p = PP[k], q = QQ[k];
            float apr = Ar[p][q], api = Ai[p][q];
            float r2 = apr * apr + api * api;
            if (r2 < 1e-26f) continue;
            float rr = sqrtf(r2);
            float car = apr / rr, sai = api / rr;              // e^{i alpha}
            float th2 = atan2f(2.f * rr, Ar[q][q] - Ar[p][p]); // 2*theta
            float th = 0.5f * th2;
            float c = cosf(th), sg = sinf(th);
            float sr = sg * car, si = sg * sai;                // s = sin(theta)*e^{i alpha}
            // column update A <- A*G  (cols p,q)
#pragma unroll
            for (int i = 0; i < 4; ++i) {
                float zpr = Ar[i][p], zpi = Ai[i][p], zqr = Ar[i][q], zqi = Ai[i][q];
                Ar[i][p] = c * zpr - (sr * zqr + si * zqi);
                Ai[i][p] = c * zpi - (sr * zqi - si * zqr);
                Ar[i][q] = (sr * zpr - si * zpi) + c * zqr;
                Ai[i][q] = (sr * zpi + si * zpr) + c * zqi;
            }
            // row update A <- G^H * A  (rows p,q)
#pragma unroll
            for (int j = 0; j < 4; ++j) {
                float zpr = Ar[p][j], zpi = Ai[p][j], zqr = Ar[q][j], zqi = Ai[q][j];
                Ar[p][j] = c * zpr - (sr * zqr - si * zqi);
                Ai[p][j] = c * zpi - (sr * zqi + si * zqr);
                Ar[q][j] = (sr * zpr + si * zpi) + c * zqr;
                Ai[q][j] = (sr * zpi - si * zpr) + c * zqi;
            }
            // eigenvector update V <- V*G
#pragma unroll
            for (int i = 0; i < 4; ++i) {
                float zpr = Vr[i][p], zpi = Vi[i][p], zqr = Vr[i][q], zqi = Vi[i][q];
                Vr[i][p] = c * zpr - (sr * zqr + si * zqi);
                Vi[i][p] = c * zpi - (sr * zqi - si * zqr);
                Vr[i][q] = (sr * zpr - si * zpi) + c * zqr;
                Vi[i][q] = (sr * zpi + si * zpr) + c * zqi;
            }
        }
    }

    // pick the 2 eigenvectors with smallest eigenvalues (noise subspace)
    float dd[4] = {Ar[0][0], Ar[1][1], Ar[2][2], Ar[3][3]};
    int idx[4] = {0, 1, 2, 3};
#pragma unroll
    for (int a = 0; a < 3; ++a)
#pragma unroll
        for (int b = 0; b < 3; ++b) {
            if (dd[b] > dd[b + 1]) {
                float tv = dd[b]; dd[b] = dd[b + 1]; dd[b + 1] = tv;
                int ti = idx[b]; idx[b] = idx[b + 1]; idx[b + 1] = ti;
            }
        }
#pragma unroll
    for (int k = 0; k < 2; ++k) {
        int ci = idx[k];
#pragma unroll
        for (int m = 0; m < 4; ++m) {
            float* o = noise + ((size_t)(f * 4 + m) * 2 + k) * 2;
            o[0] = Vr[m][ci];
            o[1] = Vi[m][ci];
        }
    }
}

// ---------------- Kernel 3: MUSIC spectrum via f32 WMMA (K=4 == mic count) ----
// grid = 23 tiles of 16 angles; block = 256 (8 waves); wave w handles f = w, w+8, ...
__global__ void __launch_bounds__(256) music_spec_kernel(const float* __restrict__ noise,
                                                         float* __restrict__ out) {
    __shared__ float lds[8 * 16];
    int tid  = threadIdx.x;
    int wv   = tid >> 5;
    int lane = tid & 31;
    int half = lane >> 4;      // 0: K rows (mics) {0,1}; 1: K rows {2,3}
    int r16  = lane & 15;
    int ab   = blockIdx.x * 16;

    int aRow = ab + r16;
    if (aRow > NANG - 1) aRow = NANG - 1;          // clamp; extra rows never stored
    float th  = (-90.0f + 0.5f * (float)aRow) * (PI_F / 180.0f);
    float sth = sinf(th);
    int   mb0 = half * 2;
    int   mb1 = mb0 + 1;
    int   n   = r16;

    float acc[8] = {0.f, 0.f, 0.f, 0.f, 0.f, 0.f, 0.f, 0.f};

    for (int f = wv; f < NFREQ; f += 8) {
        // base per-mic phase; mic phases are exact multiples -> one sincos + identities
        float p1 = KPH * sth * (FSTEP * (float)f);
        float s1, c1;
        sincosf(p1, &s1, &c1);
        float c2 = fmaf(2.f * c1, c1, -1.f);     // cos 2x
        float s2 = 2.f * s1 * c1;                // sin 2x
        float c3 = c2 * c1 - s2 * s1;            // cos 3x
        float s3 = s2 * c1 + c2 * s1;            // sin 3x
        // steer = exp(-i*m*p1): Re=cos(m p1), Im=-sin(m p1)
        v2f aSr = half ? (v2f){c2, c3} : (v2f){1.f, c1};
        v2f aSi = half ? (v2f){-s2, -s3} : (v2f){0.f, -s1};

        const float* nz = noise + (size_t)f * 16;
        float nr00 = nz[(mb0 * 2 + 0) * 2 + 0], ni00 = nz[(mb0 * 2 + 0) * 2 + 1];
        float nr01 = nz[(mb0 * 2 + 1) * 2 + 0], ni01 = nz[(mb0 * 2 + 1) * 2 + 1];
        float nr10 = nz[(mb1 * 2 + 0) * 2 + 0], ni10 = nz[(mb1 * 2 + 0) * 2 + 1];
        float nr11 = nz[(mb1 * 2 + 1) * 2 + 0], ni11 = nz[(mb1 * 2 + 1) * 2 + 1];

        // B1 cols: [nr(:,0), nr(:,1), ni(:,0), ni(:,1), 0...]
        float b1a = (n == 0) ? nr00 : (n == 1) ? nr01 : (n == 2) ? ni00 : (n == 3) ? ni01 : 0.f;
        float b1b = (n == 0) ? nr10 : (n == 1) ? nr11 : (n == 2) ? ni10 : (n == 3) ? ni11 : 0.f;
        // B2 cols: [-ni(:,0), -ni(:,1), nr(:,0), nr(:,1), 0...]
        float b2a = (n == 0) ? -ni00 : (n == 1) ? -ni01 : (n == 2) ? nr00 : (n == 3) ? nr01 : 0.f;
        float b2b = (n == 0) ? -ni10 : (n == 1) ? -ni11 : (n == 2) ? nr10 : (n == 3) ? nr11 : 0.f;
        v2f b1 = {b1a, b1b};
        v2f b2 = {b2a, b2b};

        v8f c = {0.f, 0.f, 0.f, 0.f, 0.f, 0.f, 0.f, 0.f};
        // D = Sr*B1;  D += Si*B2  => cols {0,1}=Re(mid_k), {2,3}=Im(mid_k)
        c = __builtin_amdgcn_wmma_f32_16x16x4_f32(false, aSr, false, b1, (short)0, c, false, false);
        c = __builtin_amdgcn_wmma_f32_16x16x4_f32(false, aSi, false, b2, (short)0, c, false, false);

        // row sums only need cols 0..3 (cols >= 4 are exact zeros):
        // xor-reduce within aligned 4-lane groups; lanes with r16==0 hold the sums.
#pragma unroll
        for (int r = 0; r < 8; ++r) {
            float q = c[r] * c[r];
            q += __shfl_xor(q, 1);
            q += __shfl_xor(q, 2);
            acc[r] += 1.0f / (q + MEPS);
        }
    }

    if (r16 == 0) {                     // lane 0: rows 0..7, lane 16: rows 8..15
#pragma unroll
        for (int r = 0; r < 8; ++r) lds[wv * 16 + half * 8 + r] = acc[r];
    }
    __syncthreads();
    if (tid < 16) {
        float t = 0.f;
#pragma unroll
        for (int w = 0; w < 8; ++w) t += lds[w * 16 + tid];   // fixed order: deterministic
        int a = ab + tid;
        if (a < NANG) out[a] = t * (1.0f / (float)NFREQ);
    }
}

extern "C" void kernel_launch(void* const* d_in, const int* in_sizes, int n_in,
                              void* d_out, int out_size, void* d_ws, size_t ws_size,
                              hipStream_t stream) {
    const float* x = (const float*)d_in[0];
    float* out = (float*)d_out;
    float* ws = (float*)d_ws;

    float* Xs    = ws;            // 4*257*4*2   = 8224 floats
    float* noise = ws + 8224;     // 257*4*2*2   = 4112 floats

    music_stft_kernel<<<(NFREQ * 16 + 127) / 128, 128, 0, stream>>>(x, Xs);
    music_eig_kernel<<<(NFREQ + 63) / 64, 64, 0, stream>>>(Xs, noise);
    music_spec_kernel<<<(NANG + 15) / 16, 256, 0, stream>>>(noise, out);
}